// QuantumLayer_52630529245565
// MI455X (gfx1250) — compile-verified
//
#include <hip/hip_runtime.h>
#include <hip/hip_bf16.h>
#include <math.h>

#define NQ   10
#define DIM  1024          // 2^NQ state size
#define KCH  256           // K chunk held in LDS
#define NCHK (DIM / KCH)   // 4 chunks
#define NT   32            // batch columns per block

typedef __attribute__((ext_vector_type(2))) float v2f;
typedef __attribute__((ext_vector_type(8))) float v8f;

// ---------------------------------------------------------------------------
// Phase 1: build the constant 1024x1024 unitary C = CNOTs * Rot * CRXs.
// One block per basis column j; state held in LDS; writes column j of C.
// ---------------------------------------------------------------------------
__global__ __launch_bounds__(256)
void build_C(const float* __restrict__ W, float* __restrict__ Cre,
             float* __restrict__ Cim) {
    __shared__ float sre[DIM];
    __shared__ float sim[DIM];
    const int tid = threadIdx.x;
    const int j   = blockIdx.x;

    for (int x = tid; x < DIM; x += 256) {
        sre[x] = (x == j) ? 1.0f : 0.0f;
        sim[x] = 0.0f;
    }
    __syncthreads();

    // ---- CRX chain, angle pi/3: c2 = cos(pi/6), s2 = sin(pi/6) ----
    const float c2 = 0.8660254037844387f, s2 = 0.5f;
    for (int e = 0; e < NQ; ++e) {
        const int a = e, b = (e + 1) % NQ;
        const int bpA = 9 - a, bpB = 9 - b;
        for (int p = tid; p < DIM / 2; p += 256) {
            int i0 = ((p >> bpB) << (bpB + 1)) | (p & ((1 << bpB) - 1));
            if ((i0 >> bpA) & 1) {
                int i1 = i0 | (1 << bpB);
                float r0 = sre[i0], q0 = sim[i0];
                float r1 = sre[i1], q1 = sim[i1];
                // [[c2, -i s2], [-i s2, c2]] on (t0, t1)
                sre[i0] =  c2 * r0 + s2 * q1;
                sim[i0] =  c2 * q0 - s2 * r1;
                sre[i1] =  s2 * q0 + c2 * r1;
                sim[i1] = -s2 * r0 + c2 * q1;
            }
        }
        __syncthreads();
    }

    // ---- Rot(phi, theta, omega) per wire (constant weights) ----
    for (int w = 0; w < NQ; ++w) {
        const float phi = W[w * 3 + 0], th = W[w * 3 + 1], om = W[w * 3 + 2];
        const float ct  = cosf(0.5f * th),         st  = sinf(0.5f * th);
        const float cpo = cosf(0.5f * (phi + om)), spo = sinf(0.5f * (phi + om));
        const float cpm = cosf(0.5f * (phi - om)), spm = sinf(0.5f * (phi - om));
        const float u00r =  cpo * ct, u00i = -spo * ct;
        const float u01r = -cpm * st, u01i = -spm * st;
        const float u10r =  cpm * st, u10i = -spm * st;
        const float u11r =  cpo * ct, u11i =  spo * ct;
        const int bp = 9 - w;
        for (int p = tid; p < DIM / 2; p += 256) {
            int i0 = ((p >> bp) << (bp + 1)) | (p & ((1 << bp) - 1));
            int i1 = i0 | (1 << bp);
            float r0 = sre[i0], q0 = sim[i0];
            float r1 = sre[i1], q1 = sim[i1];
            sre[i0] = u00r * r0 - u00i * q0 + u01r * r1 - u01i * q1;
            sim[i0] = u00r * q0 + u00i * r0 + u01r * q1 + u01i * r1;
            sre[i1] = u10r * r0 - u10i * q0 + u11r * r1 - u11i * q1;
            sim[i1] = u10r * q0 + u10i * r0 + u11r * q1 + u11i * r1;
        }
        __syncthreads();
    }

    // ---- CNOT(i, (i+2) % 10) layer: permutation ----
    for (int i = 0; i < NQ; ++i) {
        const int a = i, b = (i + 2) % NQ;
        const int bpA = 9 - a, bpB = 9 - b;
        for (int p = tid; p < DIM / 2; p += 256) {
            int i0 = ((p >> bpB) << (bpB + 1)) | (p & ((1 << bpB) - 1));
            if ((i0 >> bpA) & 1) {
                int i1 = i0 | (1 << bpB);
                float r0 = sre[i0], q0 = sim[i0];
                sre[i0] = sre[i1]; sim[i0] = sim[i1];
                sre[i1] = r0;      sim[i1] = q0;
            }
        }
        __syncthreads();
    }

    // ---- write column j (row-major C so GEMM A-loads are K-contiguous) ----
    for (int m = tid; m < DIM; m += 256) {
        Cre[(size_t)m * DIM + j] = sre[m];
        Cim[(size_t)m * DIM + j] = sim[m];
    }
}

// ---------------------------------------------------------------------------
// Phase 2: fused GEMM (D = C * Psi0, complex) + |D|^2 + signed-Z reduction.
// Block = 256 threads (8 wave32); block owns NT=32 batch columns.
// Each wave owns a UNIQUE 32-row M-group and BOTH column halves:
// per k-step 4 A b64 loads + 2 B ds_b64 loads feed 8 independent f32-WMMA
// chains; C matrices are read exactly once per block from L2.
// Psi0 is stored K-pair-interleaved in LDS so each B fragment is one
// contiguous 8-byte load straight into an even VGPR pair (no repack movs).
// Fully deterministic epilogue (register partials + fixed-order reduction).
// ---------------------------------------------------------------------------
// Psi0 element (kk, n) lives at  (kk>>1)*2*NT + n*2 + (kk&1).
#define PIDX(kk, n) ((((kk) >> 1) * (2 * NT)) + ((n) << 1) + ((kk) & 1))

__global__ __launch_bounds__(256)
void gemm_z(const float* __restrict__ inp, const float* __restrict__ Cre,
            const float* __restrict__ Cim, float* __restrict__ out) {
    __shared__ float psi[KCH * NT];            // 32 KB: Psi0 K-chunk
    __shared__ float csc[NT][NQ];              // cos(theta/2) per column/wire
    __shared__ float css[NT][NQ];              // sin(theta/2)
    __shared__ float zbuf[8 * 32 * 2 * NQ];    // per-(wave,lane,half) partials

    const int tid = threadIdx.x;
    const int b0  = blockIdx.x * NT;

    for (int e = tid; e < NT * NQ; e += 256) {
        int n = e / NQ, w = e % NQ;
        float th = 0.5f * inp[(size_t)(b0 + n) * NQ + w];
        csc[n][w] = cosf(th);
        css[n][w] = sinf(th);
    }

    const int lane = tid & 31;
    const int wv   = tid >> 5;      // M group 0..7
    const int nl   = lane & 15;
    const int kh   = lane >> 4;     // 0/1: K sub-pair per WMMA A/B layout

    float z[2][NQ];                 // Z partials for col nl (half0) / nl+16
#pragma unroll
    for (int hh = 0; hh < 2; ++hh)
#pragma unroll
        for (int i = 0; i < NQ; ++i) z[hh][i] = 0.0f;

    for (int mt = 0; mt < DIM / 256; ++mt) {      // 4 passes over M
        const int m0 = (mt * 8 + wv) * 32;        // wave owns rows m0..m0+31
        v8f acc[2][2][2];                         // [tile][half][re/im]
#pragma unroll
        for (int t = 0; t < 2; ++t)
#pragma unroll
            for (int hh = 0; hh < 2; ++hh)
#pragma unroll
                for (int c = 0; c < 2; ++c) acc[t][hh][c] = (v8f){};

        for (int ch = 0; ch < NCHK; ++ch) {
            const int k0 = ch * KCH;
            __syncthreads();                       // prev chunk fully consumed
            // Build Psi0 chunk (K-pair interleaved):
            // psi[PIDX(kk,n)] = prod_w (bit(x) ? sin : cos)
            for (int e = tid; e < KCH * NT; e += 256) {
                int kk = e >> 5, n = e & 31;
                int x  = k0 + kk;
                float p = 1.0f;
#pragma unroll
                for (int w = 0; w < NQ; ++w)
                    p *= ((x >> (9 - w)) & 1) ? css[n][w] : csc[n][w];
                psi[PIDX(kk, n)] = p;
            }
            __syncthreads();

            const float* Ar0 = Cre + (size_t)(m0 + nl)      * DIM + k0 + kh * 2;
            const float* Ai0 = Cim + (size_t)(m0 + nl)      * DIM + k0 + kh * 2;
            const float* Ar1 = Cre + (size_t)(m0 + 16 + nl) * DIM + k0 + kh * 2;
            const float* Ai1 = Cim + (size_t)(m0 + 16 + nl) * DIM + k0 + kh * 2;
            // B fragment base: K-pair (kh) for this lane half, column nl.
            const float* Bp  = psi + (size_t)kh * (2 * NT) + nl * 2;
            if (ch + 1 < NCHK) {                   // hint next A chunk into L2
                __builtin_prefetch(Ar0 + KCH, 0, 0);
                __builtin_prefetch(Ai0 + KCH, 0, 0);
                __builtin_prefetch(Ar1 + KCH, 0, 0);
                __builtin_prefetch(Ai1 + KCH, 0, 0);
            }
#pragma unroll 4
            for (int k = 0; k < KCH; k += 4) {
                v2f ar0 = *(const v2f*)(Ar0 + k);
                v2f ai0 = *(const v2f*)(Ai0 + k);
                v2f ar1 = *(const v2f*)(Ar1 + k);
                v2f ai1 = *(const v2f*)(Ai1 + k);
                // K-pairs (k/2 + kh); cols 0-15 at +0, cols 16-31 at +32.
                const float* bk = Bp + (size_t)(k >> 1) * (2 * NT);
                v2f b0f = *(const v2f*)(bk);
                v2f b1f = *(const v2f*)(bk + 32);
                acc[0][0][0] = __builtin_amdgcn_wmma_f32_16x16x4_f32(
                    false, ar0, false, b0f, (short)0, acc[0][0][0], false, false);
                acc[0][0][1] = __builtin_amdgcn_wmma_f32_16x16x4_f32(
                    false, ai0, false, b0f, (short)0, acc[0][0][1], false, false);
                acc[0][1][0] = __builtin_amdgcn_wmma_f32_16x16x4_f32(
                    false, ar0, false, b1f, (short)0, acc[0][1][0], false, false);
                acc[0][1][1] = __builtin_amdgcn_wmma_f32_16x16x4_f32(
                    false, ai0, false, b1f, (short)0, acc[0][1][1], false, false);
                acc[1][0][0] = __builtin_amdgcn_wmma_f32_16x16x4_f32(
                    false, ar1, false, b0f, (short)0, acc[1][0][0], false, false);
                acc[1][0][1] = __builtin_amdgcn_wmma_f32_16x16x4_f32(
                    false, ai1, false, b0f, (short)0, acc[1][0][1], false, false);
                acc[1][1][0] = __builtin_amdgcn_wmma_f32_16x16x4_f32(
                    false, ar1, false, b1f, (short)0, acc[1][1][0], false, false);
                acc[1][1][1] = __builtin_amdgcn_wmma_f32_16x16x4_f32(
                    false, ai1, false, b1f, (short)0, acc[1][1][1], false, false);
            }
        }

        // Probs + signed-Z accumulation into registers (deterministic).
        // D layout: VGPR r, lanes<16 -> M=m0t+r, lanes>=16 -> M=m0t+8+r.
        // Row-in-tile bits {lane>=16, r[2:0]} = M bits 3..0 = wires 6..9;
        // wires 0..5 have constant sign per 16-row tile (m0t bits 9..4).
#pragma unroll
        for (int t = 0; t < 2; ++t) {
            const int m0t = m0 + t * 16;
#pragma unroll
            for (int hh = 0; hh < 2; ++hh) {
                const v8f& dr = acc[t][hh][0];
                const v8f& di = acc[t][hh][1];
                float p[8];
#pragma unroll
                for (int r = 0; r < 8; ++r)
                    p[r] = dr[r] * dr[r] + di[r] * di[r];
                float sa = 0.f, s7 = 0.f, s8 = 0.f, s9 = 0.f;
#pragma unroll
                for (int r = 0; r < 8; ++r) {
                    sa += p[r];
                    s7 += (r & 4) ? -p[r] : p[r];
                    s8 += (r & 2) ? -p[r] : p[r];
                    s9 += (r & 1) ? -p[r] : p[r];
                }
#pragma unroll
                for (int i = 0; i < 6; ++i)
                    z[hh][i] += ((m0t >> (9 - i)) & 1) ? -sa : sa;
                z[hh][6] += (lane & 16) ? -sa : sa;
                z[hh][7] += s7;
                z[hh][8] += s8;
                z[hh][9] += s9;
            }
        }
    }

    // Spill per-lane partials; fixed-order reduction => deterministic output.
#pragma unroll
    for (int hh = 0; hh < 2; ++hh)
#pragma unroll
        for (int i = 0; i < NQ; ++i)
            zbuf[((wv * 32 + lane) * 2 + hh) * NQ + i] = z[hh][i];
    __syncthreads();

    for (int e = tid; e < NQ * NT; e += 256) {
        int i = e / NT, c = e % NT;
        int hh = c >> 4, nl2 = c & 15;
        float acc2 = 0.0f;
#pragma unroll
        for (int wv2 = 0; wv2 < 8; ++wv2) {
#pragma unroll
            for (int lh = 0; lh < 2; ++lh) {
                int srcl = nl2 + lh * 16;
                acc2 += zbuf[((wv2 * 32 + srcl) * 2 + hh) * NQ + i];
            }
        }
        out[(size_t)(b0 + c) * NQ + i] = acc2;
    }
}

// ---------------------------------------------------------------------------
extern "C" void kernel_launch(void* const* d_in, const int* in_sizes, int n_in,
                              void* d_out, int out_size, void* d_ws,
                              size_t ws_size, hipStream_t stream) {
    const float* inputs  = (const float*)d_in[0];   // (B, 10) f32
    const float* weights = (const float*)d_in[1];   // (1, 10, 3) f32
    float* outp = (float*)d_out;                    // (B, 10) f32

    float* Cre = (float*)d_ws;                      // 4 MB
    float* Cim = Cre + (size_t)DIM * DIM;           // 4 MB

    const int B = in_sizes[0] / NQ;                 // 8192

    build_C<<<DIM, 256, 0, stream>>>(weights, Cre, Cim);
    gemm_z<<<B / NT, 256, 0, stream>>>(inputs, Cre, Cim, outp);
}